// MDCA_15410342658796
// MI455X (gfx1250) — compile-verified
//
#include <hip/hip_runtime.h>
#include <math.h>

typedef __attribute__((ext_vector_type(2))) float v2f;
typedef __attribute__((ext_vector_type(8))) float v8f;

#define NBLK 512      // partial-sum rows (blocks in the streaming kernel)
#define CPAD 1008     // padded class count (63 chunks of 16)

__device__ __forceinline__ float waveSum(float v) {
#pragma unroll
  for (int off = 16; off > 0; off >>= 1) v += __shfl_xor(v, off, 32);
  return v;
}
__device__ __forceinline__ float waveMax(float v) {
#pragma unroll
  for (int off = 16; off > 0; off >>= 1) v = fmaxf(v, __shfl_xor(v, off, 32));
  return v;
}

// ---------------- kernel 0: zero the count buffer ----------------
__global__ void mdca_zero_counts(unsigned* __restrict__ counts) {
  if (threadIdx.x < CPAD) counts[threadIdx.x] = 0u;
}

// ---------------- kernel 1: histogram of targets (int atomics: deterministic)
__global__ void mdca_count(const int* __restrict__ target,
                           unsigned* __restrict__ counts, int B) {
  int i = blockIdx.x * blockDim.x + threadIdx.x;
  if (i < B) atomicAdd(&counts[target[i]], 1u);
}

// ---------------- kernel 2: streaming row softmax + per-block class partials
// One wave32 per row; row (<=1024 floats) lives in 32 VGPRs per lane.
// Per-lane register accumulator -> per-wave LDS row -> per-block global partial.
__global__ __launch_bounds__(256) void mdca_rows(const float* __restrict__ xin,
                                                 float* __restrict__ partials,
                                                 int B, int C) {
  const int lane  = threadIdx.x & 31;
  const int wave  = threadIdx.x >> 5;
  const int nwav  = gridDim.x * 8;
  const int gwave = blockIdx.x * 8 + wave;

  float acc[32];
#pragma unroll
  for (int t = 0; t < 32; ++t) acc[t] = 0.f;

  for (int row = gwave; row < B; row += nwav) {
    const float* rp = xin + (size_t)row * C;
    float x[32];
    float ss = 0.f;
#pragma unroll
    for (int j = 0; j < 8; ++j) {             // float4 coalesced: 512B/iter/wave
      int e0 = 4 * (lane + 32 * j);
      float v[4];
      if (e0 + 4 <= C) {
        float4 q = *(const float4*)(rp + e0);
        v[0] = q.x; v[1] = q.y; v[2] = q.z; v[3] = q.w;
      } else {
#pragma unroll
        for (int k = 0; k < 4; ++k) v[k] = (e0 + k < C) ? rp[e0 + k] : 0.f;
      }
#pragma unroll
      for (int k = 0; k < 4; ++k) {
        bool valid = (e0 + k) < C;
        float xv = valid ? v[k] : -__builtin_inff();
        x[4 * j + k] = xv;
        ss += valid ? xv * xv : 0.f;
      }
    }
    ss = waveSum(ss);
    float rn = 1.f / (sqrtf(ss) + 1e-7f);      // 1/(||x||+eps)
    float m = -__builtin_inff();
#pragma unroll
    for (int t = 0; t < 32; ++t) { x[t] *= rn; m = fmaxf(m, x[t]); }
    m = waveMax(m);
    float s = 0.f;
#pragma unroll
    for (int t = 0; t < 32; ++t) { x[t] = expf(x[t] - m); s += x[t]; }
    s = waveSum(s);
    float inv = 1.f / s;
#pragma unroll
    for (int t = 0; t < 32; ++t) acc[t] += x[t] * inv;   // per-class running sum
  }

  // merge 8 wave accumulators -> one per-block partial row (deterministic order)
  __shared__ float lacc[8][CPAD];
#pragma unroll
  for (int t = 0; t < 32; ++t) {
    int c = 4 * lane + 128 * (t >> 2) + (t & 3);   // class owned by (lane,t); bijective on [0,1024)
    if (c < CPAD) lacc[wave][c] = acc[t];
  }
  __syncthreads();
  for (int c = threadIdx.x; c < CPAD; c += 256) {
    float s = 0.f;
#pragma unroll
    for (int w = 0; w < 8; ++w) s += lacc[w][c];
    partials[(size_t)blockIdx.x * CPAD + c] = s;
  }
}

// ---------------- kernel 3: WMMA column reduction + final scalar
// avg_conf = ones(1,512) x partials(512x1008): fold 4 rows/chunk per
// V_WMMA_F32_16X16X4_F32 with an all-ones A. All-ones A makes the result
// invariant to B's K-row ordering; N maps to lane&15 as in the C/D layout.
__global__ __launch_bounds__(1024) void mdca_finalize(const float* __restrict__ partials,
                                                      const unsigned* __restrict__ counts,
                                                      float* __restrict__ out,
                                                      int B, int C) {
  __shared__ float csum[64];
  const int lane = threadIdx.x & 31;
  const int wave = threadIdx.x >> 5;
  if (threadIdx.x < 64) csum[threadIdx.x] = 0.f;
  __syncthreads();

  const int half = lane >> 4;     // 0: K=0/1 rows, 1: K=2/3 rows
  const int n    = lane & 15;     // class-within-chunk (N)
  v2f a; a.x = 1.0f; a.y = 1.0f;  // all-ones A (16x4)

  for (int chunk = wave; chunk < CPAD / 16; chunk += 32) {
    const float* colbase = partials + chunk * 16 + n;
    v8f d = {};
    for (int r0 = 0; r0 < NBLK; r0 += 4) {
      v2f b;
      b.x = colbase[(size_t)(r0 + 2 * half) * CPAD];
      b.y = colbase[(size_t)(r0 + 1 + 2 * half) * CPAD];
      d = __builtin_amdgcn_wmma_f32_16x16x4_f32(false, a, false, b,
                                                (short)0, d, false, false);
    }
    // d[0]: lanes 0-15 hold M=0,N=lane (column sums, replicated over M)
    int c = chunk * 16 + n;
    float term = 0.f;
    if (half == 0 && c < C) term = fabsf(d[0] - (float)counts[c]);
    term = waveSum(term);                       // lanes 16-31 contribute 0
    if (lane == 0) csum[chunk] = term;
  }
  __syncthreads();
  if (threadIdx.x == 0) {
    float t = 0.f;
    for (int i = 0; i < CPAD / 16; ++i) t += csum[i];   // fixed order: deterministic
    out[0] = t / ((float)B * (float)C);                 // /B for both means, /C for class mean
  }
}

extern "C" void kernel_launch(void* const* d_in, const int* in_sizes, int n_in,
                              void* d_out, int out_size, void* d_ws, size_t ws_size,
                              hipStream_t stream) {
  const float* output = (const float*)d_in[0];
  const int*   target = (const int*)d_in[1];
  const int B = in_sizes[1];
  const int C = in_sizes[0] / B;   // 1000 (design limit: C <= CPAD = 1008)

  unsigned* counts   = (unsigned*)d_ws;                       // CPAD uints
  float*    partials = (float*)((char*)d_ws + 4096);          // NBLK x CPAD floats (~2MB)
  float*    out      = (float*)d_out;

  mdca_zero_counts<<<1, 1024, 0, stream>>>(counts);
  mdca_count<<<(B + 255) / 256, 256, 0, stream>>>(target, counts, B);
  mdca_rows<<<NBLK, 256, 0, stream>>>(output, partials, B, C);
  mdca_finalize<<<1, 1024, 0, stream>>>(partials, counts, out, B, C);
}